// LightGCN_7129645711633
// MI455X (gfx1250) — compile-verified
//
#include <hip/hip_runtime.h>

#define N_SNO_    120000
#define N_DIS_    80000
#define N_NODES_  200000
#define NNZ_      3200000
#define DIM_      128
#define BATCH_    16384
#define N_LAYERS_ 3

typedef __attribute__((ext_vector_type(2))) float v2f;
typedef __attribute__((ext_vector_type(8))) float v8f;

__device__ __forceinline__ void atomic_add_f32(float* p, float v) {
  // relaxed, device scope -> native global_atomic_add_f32 (no CAS loop)
  (void)__hip_atomic_fetch_add(p, v, __ATOMIC_RELAXED, __HIP_MEMORY_SCOPE_AGENT);
}

// ---------------------------------------------------------------------------
// 1) x0 = concat(emb_sno, emb_dis), float4 wide
// ---------------------------------------------------------------------------
__global__ void k_init_x(const float* __restrict__ es, const float* __restrict__ ed,
                         float* __restrict__ x) {
  size_t t = (size_t)blockIdx.x * blockDim.x + threadIdx.x;
  size_t node = t >> 5;
  int q = (int)(t & 31);
  if (node >= (size_t)N_NODES_) return;
  const float* srcp = (node < (size_t)N_SNO_) ? (es + node * DIM_)
                                              : (ed + (node - N_SNO_) * DIM_);
  *(float4*)(x + node * DIM_ + q * 4) = *(const float4*)(srcp + q * 4);
}

// ---------------------------------------------------------------------------
// 2) Batch-row accumulators: accS[b] (+)= x[sno[b]], accT[b] (+)= x[N_SNO+dis[b]]
// ---------------------------------------------------------------------------
__global__ void k_gather_acc(const float* __restrict__ x,
                             const int* __restrict__ sno,
                             const int* __restrict__ dis,
                             float* __restrict__ accS, float* __restrict__ accT,
                             int add) {
  int t = blockIdx.x * blockDim.x + threadIdx.x;
  int b = t >> 5, q = t & 31;
  if (b >= BATCH_) return;
  size_t rs = (size_t)sno[b] * DIM_ + q * 4;
  size_t rt = ((size_t)N_SNO_ + (size_t)dis[b]) * DIM_ + q * 4;
  float4 s  = *(const float4*)(x + rs);
  float4 tt = *(const float4*)(x + rt);
  float4* ps = (float4*)(accS + (size_t)b * DIM_ + q * 4);
  float4* pt = (float4*)(accT + (size_t)b * DIM_ + q * 4);
  if (add) {
    float4 a = *ps; a.x += s.x;  a.y += s.y;  a.z += s.z;  a.w += s.w;  *ps = a;
    float4 c = *pt; c.x += tt.x; c.y += tt.y; c.z += tt.z; c.w += tt.w; *pt = c;
  } else {
    *ps = s; *pt = tt;
  }
}

// ---------------------------------------------------------------------------
// 3) COO SpMM scatter: y[dst] += val * x[src]
//    Wave loads 32 edges coalesced with NT hint (edge stream must not evict
//    the ~200MB node tables from the 192MB L2), then broadcasts each edge.
// ---------------------------------------------------------------------------
__global__ void k_spmm(const float* __restrict__ x, float* __restrict__ y,
                       const int* __restrict__ esrc, const int* __restrict__ edst,
                       const float* __restrict__ eval) {
  int lane   = threadIdx.x & 31;
  int wave   = (blockIdx.x * blockDim.x + threadIdx.x) >> 5;
  int nwaves = (gridDim.x * blockDim.x) >> 5;
  for (int base = wave * 32; base < NNZ_; base += nwaves * 32) {
    int n = NNZ_ - base; if (n > 32) n = 32;
    int s = 0, d = 0; float v = 0.f;
    if (lane < n) {
      s = __builtin_nontemporal_load(esrc + base + lane);
      d = __builtin_nontemporal_load(edst + base + lane);
      v = __builtin_nontemporal_load(eval + base + lane);
    }
    for (int j = 0; j < n; ++j) {
      int   sj = __shfl(s, j);
      int   dj = __shfl(d, j);
      float vj = __shfl(v, j);
      const float4 xs = *(const float4*)(x + (size_t)sj * DIM_ + lane * 4);
      float* yp = y + (size_t)dj * DIM_ + lane * 4;
      atomic_add_f32(yp + 0, xs.x * vj);
      atomic_add_f32(yp + 1, xs.y * vj);
      atomic_add_f32(yp + 2, xs.z * vj);
      atomic_add_f32(yp + 3, xs.w * vj);
    }
  }
}

// ---------------------------------------------------------------------------
// 4) gamma[b] = (1/16) * dot(accS[b], accT[b]) via V_WMMA_F32_16X16X4_F32.
//    One wave computes the 16x16 tile S_tile x T_tile^T over K=128 (32 WMMAs)
//    and extracts the diagonal.
//    A (16x4 f32, 2 VGPR): lanes 0-15 M=lane {K=k0,k0+1}; lanes 16-31 {K=k0+2,k0+3}.
//    B (4x16 f32, 2 VGPR): lanes 0-15 N=lane {K=k0,k0+1}; lanes 16-31 {K=k0+2,k0+3}.
//    C/D diag: m=0..7 -> lane m, vgpr m ; m=8..15 -> lane m+16, vgpr m-8.
// ---------------------------------------------------------------------------
__global__ void k_dot_wmma(const float* __restrict__ S, const float* __restrict__ T,
                           float* __restrict__ out) {
  int lane = threadIdx.x & 31;
  int wave = (blockIdx.x * blockDim.x + threadIdx.x) >> 5;
  int b0 = wave * 16;
  if (b0 >= BATCH_) return;                 // wave-uniform: EXEC stays all-1s
  int row   = b0 + (lane & 15);
  int kbase = (lane >> 4) * 2;              // 0 for lanes 0-15, 2 for lanes 16-31
  const float* sp = S + (size_t)row * DIM_ + kbase;
  const float* tp = T + (size_t)row * DIM_ + kbase;
  v8f c = {};
#pragma unroll
  for (int k0 = 0; k0 < DIM_; k0 += 4) {
    v2f a, b;
    a.x = sp[k0]; a.y = sp[k0 + 1];         // contiguous -> global_load_b64
    b.x = tp[k0]; b.y = tp[k0 + 1];
    c = __builtin_amdgcn_wmma_f32_16x16x4_f32(
        /*neg_a=*/false, a, /*neg_b=*/false, b,
        /*c_mod=*/(short)0, c, /*reuse_a=*/false, /*reuse_b=*/false);
  }
  // diagonal extraction
  int idx = (lane < 8) ? lane : (lane - 24);  // meaningful for lanes 0-7, 24-31
  float v = 0.f;
#pragma unroll
  for (int r = 0; r < 8; ++r) v = (idx == r) ? c[r] : v;
  if (lane < 8) {
    out[b0 + lane] = v * 0.0625f;             // 1/16 = (1/4)^2
  } else if (lane >= 24) {
    out[b0 + (lane - 16)] = v * 0.0625f;
  }
}

// ---------------------------------------------------------------------------
extern "C" void kernel_launch(void* const* d_in, const int* in_sizes, int n_in,
                              void* d_out, int out_size, void* d_ws, size_t ws_size,
                              hipStream_t stream) {
  const float* emb_sno  = (const float*)d_in[0];
  const float* emb_dis  = (const float*)d_in[1];
  const float* edge_val = (const float*)d_in[2];
  const int*   snoRNAs  = (const int*)d_in[3];
  const int*   diseases = (const int*)d_in[4];
  const int*   edge_src = (const int*)d_in[5];
  const int*   edge_dst = (const int*)d_in[6];
  float* out = (float*)d_out;

  const size_t node_elems = (size_t)N_NODES_ * DIM_;   // 25.6M floats, 102.4 MB
  float* x0   = (float*)d_ws;
  float* x1   = x0 + node_elems;
  float* accS = x1 + node_elems;
  float* accT = accS + (size_t)BATCH_ * DIM_;

  // x0 = concat(emb_sno, emb_dis); acc = layer-0 contribution
  k_init_x<<<(N_NODES_ * 32) / 256, 256, 0, stream>>>(emb_sno, emb_dis, x0);
  k_gather_acc<<<(BATCH_ * 32) / 256, 256, 0, stream>>>(x0, snoRNAs, diseases,
                                                        accS, accT, /*add=*/0);

  float* cur = x0;
  float* nxt = x1;
  for (int l = 0; l < N_LAYERS_; ++l) {
    hipMemsetAsync(nxt, 0, node_elems * sizeof(float), stream);
    k_spmm<<<8192, 256, 0, stream>>>(cur, nxt, edge_src, edge_dst, edge_val);
    k_gather_acc<<<(BATCH_ * 32) / 256, 256, 0, stream>>>(nxt, snoRNAs, diseases,
                                                          accS, accT, /*add=*/1);
    float* t = cur; cur = nxt; nxt = t;
  }

  // gamma[b] = (1/16) * <accS[b], accT[b]>
  k_dot_wmma<<<(BATCH_ / 16) * 32 / 256, 256, 0, stream>>>(accS, accT, out);
}